// QLSTM_65481071398573
// MI455X (gfx1250) — compile-verified
//
#include <hip/hip_runtime.h>
#include <math.h>

#define NQ 10
#define DEPTH 4
#define SDIM 1024
#define BDIM 256
#define HXP_STRIDE 12   // K padded 10 -> 12 for three 16x16x4 WMMA steps

// ---- workspace layout (float offsets) ----
#define WS_GATES 0        // 40 floats: f[10] | i[10] | g[10] | o[10]  (activated)
#define WS_HXF   40       // 10 floats: final hx
#define WS_CXF   50       // 10 floats: final cx
#define WS_HXP   64       // 1024*12 floats: padded hx table (zeros at h=10,11)
#define WS_ROW   16384    // 1024*10 floats: row[t][d] = hx_t . post_W[d,:] + post_b[d]

typedef __attribute__((ext_vector_type(2))) float v2f;
typedef __attribute__((ext_vector_type(8))) float v8f;

struct cpx { float re, im; };
__device__ __forceinline__ cpx cmul(cpx a, cpx b) {
    return {a.re * b.re - a.im * b.im, a.re * b.im + a.im * b.re};
}
__device__ __forceinline__ cpx cadd(cpx a, cpx b) { return {a.re + b.re, a.im + b.im}; }

// ---------------------------------------------------------------------------
// Kernel 1: statevector simulation of the 4 variational circuits.
// One block per circuit; 256 threads each own 4 of the 1024 amplitudes in LDS.
// Per layer: 10 fused (RZ*RY*RX) single-qubit gates + whole CNOT chain fused
// into one permutation gather. Then per-wire <Z> via warp reductions.
// ---------------------------------------------------------------------------
__global__ __launch_bounds__(256) void qcirc_kernel(
    const float* __restrict__ p0, const float* __restrict__ p1,
    const float* __restrict__ p2, const float* __restrict__ p3,
    float* __restrict__ ws)
{
    __shared__ float sre[1024];
    __shared__ float sim_[1024];
    const int g   = blockIdx.x;                  // 0=f 1=i 2=g 3=o
    const float* P = (g == 0) ? p0 : (g == 1) ? p1 : (g == 2) ? p2 : p3;
    const int tid = threadIdx.x;

#pragma unroll
    for (int j = 0; j < 4; ++j) {
        int idx = tid + 256 * j;
        sre[idx] = (idx == 0) ? 1.f : 0.f;
        sim_[idx] = 0.f;
    }
    __syncthreads();

    for (int d = 0; d < DEPTH; ++d) {
        for (int w = 0; w < NQ; ++w) {
            // Fused U = Rz(t2) * Ry(t1) * Rx(t0)  (Rx applied first)
            float t0 = P[(d * NQ + w) * 3 + 0] * 0.5f;
            float t1 = P[(d * NQ + w) * 3 + 1] * 0.5f;
            float t2 = P[(d * NQ + w) * 3 + 2] * 0.5f;
            float c0 = cosf(t0), s0 = sinf(t0);
            float c1 = cosf(t1), s1 = sinf(t1);
            float c2 = cosf(t2), s2 = sinf(t2);
            cpx x00 = {c0, 0.f}, x01 = {0.f, -s0}, x10 = {0.f, -s0}, x11 = {c0, 0.f};
            cpx y00 = {c1, 0.f}, y01 = {-s1, 0.f}, y10 = {s1, 0.f}, y11 = {c1, 0.f};
            cpx m00 = cadd(cmul(y00, x00), cmul(y01, x10));
            cpx m01 = cadd(cmul(y00, x01), cmul(y01, x11));
            cpx m10 = cadd(cmul(y10, x00), cmul(y11, x10));
            cpx m11 = cadd(cmul(y10, x01), cmul(y11, x11));
            cpx z0 = {c2, -s2}, z1 = {c2, s2};
            cpx u00 = cmul(z0, m00), u01 = cmul(z0, m01);
            cpx u10 = cmul(z1, m10), u11 = cmul(z1, m11);

            const int mask = 1 << (9 - w);
            float nr[4], ni[4];
#pragma unroll
            for (int j = 0; j < 4; ++j) {
                int idx = tid + 256 * j;
                int i0 = idx & ~mask, i1 = idx | mask;
                cpx a0 = {sre[i0], sim_[i0]};
                cpx a1 = {sre[i1], sim_[i1]};
                bool hi = (idx & mask) != 0;
                cpx ua = hi ? u10 : u00;
                cpx ub = hi ? u11 : u01;
                cpx nv = cadd(cmul(ua, a0), cmul(ub, a1));
                nr[j] = nv.re; ni[j] = nv.im;
            }
            __syncthreads();
#pragma unroll
            for (int j = 0; j < 4; ++j) {
                int idx = tid + 256 * j;
                sre[idx] = nr[j]; sim_[idx] = ni[j];
            }
            __syncthreads();
        }
        // Fused CNOT chain: final[x] = state[g0(g1(...g8(x)))]
        {
            float nr[4], ni[4];
#pragma unroll
            for (int j = 0; j < 4; ++j) {
                int idx = tid + 256 * j;
                int src = idx;
#pragma unroll
                for (int w = 8; w >= 0; --w) {
                    if (src & (1 << (9 - w))) src ^= (1 << (8 - w));
                }
                nr[j] = sre[src]; ni[j] = sim_[src];
            }
            __syncthreads();
#pragma unroll
            for (int j = 0; j < 4; ++j) {
                int idx = tid + 256 * j;
                sre[idx] = nr[j]; sim_[idx] = ni[j];
            }
            __syncthreads();
        }
    }

    // probabilities into sre
    float p[4];
#pragma unroll
    for (int j = 0; j < 4; ++j) {
        int idx = tid + 256 * j;
        p[j] = sre[idx] * sre[idx] + sim_[idx] * sim_[idx];
    }
    __syncthreads();
#pragma unroll
    for (int j = 0; j < 4; ++j) sre[tid + 256 * j] = p[j];
    __syncthreads();

    // per-wire <Z>: warp wid handles wires wid, wid+8
    const int wid = tid >> 5, lane = tid & 31;
    for (int w = wid; w < NQ; w += 8) {
        float s = 0.f;
        for (int idx = lane; idx < 1024; idx += 32)
            s += sre[idx] * (((idx >> (9 - w)) & 1) ? -1.f : 1.f);
#pragma unroll
        for (int off = 16; off > 0; off >>= 1) s += __shfl_down(s, off, 32);
        if (lane == 0) {
            float v = (g == 2) ? tanhf(s) : 1.f / (1.f + expf(-s));
            ws[WS_GATES + g * NQ + w] = v;
        }
    }
}

// ---------------------------------------------------------------------------
// Kernel 2: closed-form recurrence, fully parallel over (t, h).
// cx_{t+1} = i*g*(1 - f^{t+1})/(1-f);  hx_t = o*tanh(cx_{t+1}).
// Writes K-padded hx table (stride 12, zeros at h>=10) + final hx/cx.
// ---------------------------------------------------------------------------
__global__ __launch_bounds__(256) void recurrence_kernel(float* __restrict__ ws)
{
    int e = blockIdx.x * blockDim.x + threadIdx.x;
    if (e >= SDIM * HXP_STRIDE) return;
    int t = e / HXP_STRIDE;
    int h = e % HXP_STRIDE;
    float hx = 0.f, cx = 0.f;
    if (h < NQ) {
        float f  = ws[WS_GATES + 0 * NQ + h];   // in (0.27, 0.73): 1-f safe
        float ii = ws[WS_GATES + 1 * NQ + h];
        float gg = ws[WS_GATES + 2 * NQ + h];
        float oo = ws[WS_GATES + 3 * NQ + h];
        float pw = expf((float)(t + 1) * logf(f));
        cx = ii * gg * (1.f - pw) / (1.f - f);
        hx = oo * tanhf(cx);
    }
    ws[WS_HXP + t * HXP_STRIDE + h] = hx;
    if (t == SDIM - 1 && h < NQ) {
        ws[WS_HXF + h] = hx;
        ws[WS_CXF + h] = cx;
    }
}

// ---------------------------------------------------------------------------
// Kernel 3: row[t][d] = hx_t . post_W[d,:] + post_b[d] via V_WMMA_F32_16X16X4_F32.
// One wave per 16-timestep tile; N=16 (10 valid), K=12 -> 3 chained wmmas.
// A[m][k]: lane = (m%16) + 16*(k/2), reg = k%2  (ISA 16x4 f32 layout)
// B[k][n]: lane = (n%16) + 16*(k/2), reg = k%2
// D[m][n]: lane = n + 16*(m/8),      reg = m%8
// ---------------------------------------------------------------------------
__global__ __launch_bounds__(32) void rowgemm_kernel(
    float* ws, const float* __restrict__ postW, const float* __restrict__ postB)
{
    const int tm   = blockIdx.x;      // 0..63
    const int lane = threadIdx.x;     // 0..31
    const int half = lane >> 4;
    const int lm   = lane & 15;
    const float* hxp = ws + WS_HXP;

    float bias = (lm < NQ) ? postB[lm] : 0.f;
    v8f c;
#pragma unroll
    for (int r = 0; r < 8; ++r) c[r] = bias;

#if defined(__has_builtin) && __has_builtin(__builtin_amdgcn_wmma_f32_16x16x4_f32)
#pragma unroll
    for (int s = 0; s < 3; ++s) {
        int k0 = 4 * s + 2 * half;
        v2f a, b;
        a.x = hxp[(tm * 16 + lm) * HXP_STRIDE + k0];        // padded: 0 at k>=10
        a.y = hxp[(tm * 16 + lm) * HXP_STRIDE + k0 + 1];
        b.x = (lm < NQ && k0 < NQ)     ? postW[lm * NQ + k0]     : 0.f;
        b.y = (lm < NQ && k0 + 1 < NQ) ? postW[lm * NQ + k0 + 1] : 0.f;
        c = __builtin_amdgcn_wmma_f32_16x16x4_f32(
                /*neg_a=*/false, a, /*neg_b=*/false, b,
                /*c_mod=*/(short)0, c, /*reuse_a=*/false, /*reuse_b=*/false);
    }
#else
    if (lm < NQ) {
#pragma unroll
        for (int r = 0; r < 8; ++r) {
            int t = tm * 16 + 8 * half + r;
            float acc = bias;
            for (int h = 0; h < NQ; ++h)
                acc += hxp[t * HXP_STRIDE + h] * postW[lm * NQ + h];
            c[r] = acc;
        }
    }
#endif

    float* row = ws + WS_ROW;
    if (lm < NQ) {
#pragma unroll
        for (int r = 0; r < 8; ++r) {
            int t = tm * 16 + 8 * half + r;
            row[t * NQ + lm] = c[r];
        }
    }
}

// ---------------------------------------------------------------------------
// Kernel 4: broadcast row[t][:] over batch with float4 stores.
// Per t-plane: 2560 floats = 640 float4; value pattern has period 10 float4.
// ---------------------------------------------------------------------------
__global__ __launch_bounds__(256) void bcast_kernel(
    const float* __restrict__ ws, float* __restrict__ out)
{
    __shared__ __align__(16) float pat[40];
    const int t   = blockIdx.x;
    const int tid = threadIdx.x;
    if (tid < 40) pat[tid] = ws[WS_ROW + t * NQ + (tid % NQ)];
    __syncthreads();
    const float4* p4   = (const float4*)pat;
    float4*       out4 = (float4*)(out + (size_t)t * BDIM * NQ);
    for (int q = tid; q < (BDIM * NQ) / 4; q += 256)
        out4[q] = p4[q % 10];
}

// ---------------------------------------------------------------------------
// Kernel 5: final (hx, cx), each (256,10), broadcast from ws.
// ---------------------------------------------------------------------------
__global__ __launch_bounds__(256) void tails_kernel(
    const float* __restrict__ ws, float* __restrict__ out)
{
    int tid = blockIdx.x * blockDim.x + threadIdx.x;
    const int n = BDIM * NQ;           // 2560
    if (tid >= 2 * n) return;
    const size_t TAIL = (size_t)SDIM * BDIM * NQ;
    int which = tid / n;               // 0=hx, 1=cx
    int h = (tid % n) % NQ;
    out[TAIL + tid] = ws[(which == 0 ? WS_HXF : WS_CXF) + h];
}

extern "C" void kernel_launch(void* const* d_in, const int* in_sizes, int n_in,
                              void* d_out, int out_size, void* d_ws, size_t ws_size,
                              hipStream_t stream)
{
    (void)in_sizes; (void)n_in; (void)out_size; (void)ws_size;
    // setup_inputs order:
    // 0 inputs (dead), 1..4 forget/input/update/output params,
    // 5..12 gate W/b (dead), 13 post_W (10x10), 14 post_b (10)
    const float* forget_params = (const float*)d_in[1];
    const float* input_params  = (const float*)d_in[2];
    const float* update_params = (const float*)d_in[3];
    const float* output_params = (const float*)d_in[4];
    const float* post_W        = (const float*)d_in[13];
    const float* post_b        = (const float*)d_in[14];
    float* ws  = (float*)d_ws;
    float* out = (float*)d_out;

    qcirc_kernel<<<4, 256, 0, stream>>>(forget_params, input_params,
                                        update_params, output_params, ws);
    recurrence_kernel<<<(SDIM * HXP_STRIDE) / 256, 256, 0, stream>>>(ws);
    rowgemm_kernel<<<SDIM / 16, 32, 0, stream>>>(ws, post_W, post_b);
    bcast_kernel<<<SDIM, 256, 0, stream>>>(ws, out);
    tails_kernel<<<(2 * BDIM * NQ + 255) / 256, 256, 0, stream>>>(ws, out);
}